// NeighborhoodNormalization_45217415692927
// MI455X (gfx1250) — compile-verified
//
#include <hip/hip_runtime.h>

typedef __attribute__((ext_vector_type(2))) float v2f;
typedef __attribute__((ext_vector_type(8))) float v8f;

// One wave32 per point. Each wave:
//   1. builds the analytic inverse of the per-point homogeneous transform
//      as a 16x4 f32 A-matrix (rows 0..2 = Tinv rows, rest zero),
//   2. loads 16 homogeneous neighbor columns into the 4x16 B-matrix
//      (branchless lane selects -> one full-wave global_load_b96 per tile),
//   3. runs V_WMMA_F32_16X16X4_F32: D rows 0..2 = transformed x/y/z,
//   4. lane n (<16) stores neighbor n's xyz (global_store_b96, NT hint).
__global__ __launch_bounds__(256)
void neigh_norm_wmma_kernel(const float* __restrict__ points,
                            const float* __restrict__ nbh,
                            float* __restrict__ out,
                            int nPoints)
{
    const int lane = threadIdx.x & 31;
    const int wave = threadIdx.x >> 5;
    const int p    = blockIdx.x * 8 + wave;   // wave-uniform
    if (p >= nPoints) return;                 // uniform exit: EXEC stays full

    // ---- per-point transform (uniform across the wave) ----
    const float px = points[3 * p + 0];
    const float py = points[3 * p + 1];
    const float pz = points[3 * p + 2];

    const float n2      = px * px + py * py + pz * pz;
    const float nrm     = sqrtf(n2);
    const float inv_nrm = (nrm == 0.0f) ? 0.0f : (1.0f / nrm);  // div_no_nan
    const float ux = px * inv_nrm;
    const float uy = py * inv_nrm;

    // T = [[ux,-uy,0,px],[uy,ux,0,py],[0,0,1,pz],[0,0,0,1]]
    // Top-left 2x2 det = ux^2+uy^2 ; analytic inverse of T:
    const float dd   = ux * ux + uy * uy;
    const float invd = (dd == 0.0f) ? 0.0f : (1.0f / dd);
    const float t00 =  ux * invd;
    const float t01 =  uy * invd;
    const float t03 = -(ux * px + uy * py) * invd;
    const float t10 = -uy * invd;
    const float t11 =  ux * invd;
    const float t13 =  (uy * px - ux * py) * invd;
    // row2 = [0, 0, 1, -pz]; row3 never needed (w output dropped).

    // ---- A matrix (16x4 f32), ISA 32-bit A layout:
    //      lane L<16 holds {A[L][0],A[L][1]}, lane L>=16 {A[L-16][2],A[L-16][3]}.
    //      Built branchlessly (v_cndmask), no exec divergence. ----
    const int  row = lane & 15;
    const bool lo  = lane < 16;
    v2f a;
    a.x = lo ? ((row == 0) ? t00 : (row == 1) ? t10 : 0.0f)
             : ((row == 2) ? 1.0f : 0.0f);
    a.y = lo ? ((row == 0) ? t01 : (row == 1) ? t11 : 0.0f)
             : ((row == 0) ? t03 : (row == 1) ? t13 : (row == 2) ? -pz : 0.0f);

    const size_t nbase = (size_t)p * 64u * 3u;
    const int    col   = lane & 15;

    #pragma unroll
    for (int t = 0; t < 4; ++t) {
        const size_t ni = nbase + (size_t)(t * 16 + col) * 3u;

        if (t < 3) {  // hint the next 16-neighbor tile (global_prefetch_b8)
            __builtin_prefetch(nbh + ni + 48, 0, 0);
        }

        // Full-wave unconditional 12B load (merges into global_load_b96).
        const float f0 = nbh[ni + 0];
        const float f1 = nbh[ni + 1];
        const float f2 = nbh[ni + 2];

        // B matrix (4x16 f32): column `col` = [nx, ny, nz, 1]^T,
        // lanes 0-15 carry rows K=0,1 ; lanes 16-31 rows K=2,3.
        // Branchless: both sides of the cndmask use loaded values.
        v2f b;
        b.x = lo ? f0 : f2;
        b.y = lo ? f1 : 1.0f;

        v8f c = {};
        // D = A x B + 0 : v_wmma_f32_16x16x4_f32 (EXEC is all-1s here)
        c = __builtin_amdgcn_wmma_f32_16x16x4_f32(
                /*neg_a=*/false, a, /*neg_b=*/false, b,
                /*c_mod=*/(short)0, c, /*reuse_a=*/false, /*reuse_b=*/false);

        // D layout: lane L<16 holds column N=L, rows M=0..7 in c[0..7].
        // Rows 0..2 are the transformed x/y/z of neighbor (t*16+L).
        if (lo) {
            float* o = out + ni;
            __builtin_nontemporal_store(c[0], o + 0);
            __builtin_nontemporal_store(c[1], o + 1);
            __builtin_nontemporal_store(c[2], o + 2);
        }
    }
}

extern "C" void kernel_launch(void* const* d_in, const int* in_sizes, int n_in,
                              void* d_out, int out_size, void* d_ws, size_t ws_size,
                              hipStream_t stream)
{
    const float* points = (const float*)d_in[0];   // [B, N, 3]
    const float* nbh    = (const float*)d_in[1];   // [B, N, 64, 3]
    float*       out    = (float*)d_out;           // [B, N, 64, 3]

    const int nPoints = in_sizes[0] / 3;           // B*N = 131072
    const int blocks  = (nPoints + 7) / 8;         // 8 waves (points) per block

    neigh_norm_wmma_kernel<<<blocks, 256, 0, stream>>>(points, nbh, out, nPoints);
}